// MultiBehavioralGNN_56899726737494
// MI455X (gfx1250) — compile-verified
//
#include <hip/hip_runtime.h>
#include <hip/hip_bf16.h>

typedef float v2f __attribute__((ext_vector_type(2)));
typedef float v8f __attribute__((ext_vector_type(8)));

#define TB 256

__device__ __forceinline__ float leaky(float x) { return x > 0.f ? x : 0.01f * x; }

__device__ __forceinline__ void atomAddF(float* p, float v) {
    (void)__hip_atomic_fetch_add(p, v, __ATOMIC_RELAXED, __HIP_MEMORY_SCOPE_AGENT);
}

// ---------------- utility kernels ----------------
__global__ void k_zero_f32(float* p, int n) {
    int i = blockIdx.x * blockDim.x + threadIdx.x;
    if (i < n) p[i] = 0.f;
}

__global__ void k_zero_u32(unsigned* p, int n) {
    int i = blockIdx.x * blockDim.x + threadIdx.x;
    if (i < n) p[i] = 0u;
}

// ---------------- degrees & norms ----------------
__global__ void k_degree(const int* __restrict__ src, const int* __restrict__ dst,
                         int E, unsigned* odeg, unsigned* ideg) {
    int e = blockIdx.x * blockDim.x + threadIdx.x;
    if (e < E) {
        atomicAdd(&odeg[src[e]], 1u);
        atomicAdd(&ideg[dst[e]], 1u);
    }
}

__global__ void k_norm(const unsigned* __restrict__ deg, float* __restrict__ nrm, int n) {
    int i = blockIdx.x * blockDim.x + threadIdx.x;
    if (i < n) {
        unsigned d = deg[i];
        nrm[i] = rsqrtf((float)(d ? d : 1u));
    }
}

// out[i][c] = src[i % srcN][c] * onorm[i] for i < rows; zeros for rows <= i < rowsPad.
// (kron-tile expansion for layer 2, plus deterministic zero-padding so the WMMA
//  GEMM below can run guard-free over whole 128-row blocks)
__global__ void k_expand_scale(const float* __restrict__ src, const float* __restrict__ onorm,
                               float* __restrict__ out, int rows, int rowsPad, int srcN) {
    int t = blockIdx.x * blockDim.x + threadIdx.x;
    int i = t >> 5;
    if (i >= rowsPad) return;
    int c = (t & 31) * 4;
    float4 x;
    if (i < rows) {
        int si = i % srcN;
        x = *(const float4*)(src + (size_t)si * 128 + c);
        float s = onorm[i];
        x.x *= s; x.y *= s; x.z *= s; x.w *= s;
    } else {
        x = make_float4(0.f, 0.f, 0.f, 0.f);
    }
    *(float4*)(out + (size_t)i * 128 + c) = x;
}

// ---------------- in-place 128-wide WMMA GEMM: buf(rowsPad x 128) = buf @ W(128x128) ----------------
// rowsPad is a multiple of 128 and pad rows are pre-zeroed, so no bounds checks anywhere.
// Block = 256 threads = 8 waves. Wave w owns rows [blk*128 + w*16, +16): it reads only those
// rows (K loop) and writes only those rows (after the K loop), so in-place is race-free.
__global__ void __launch_bounds__(TB) k_gemm128_inplace(float* __restrict__ buf,
                                                        const float* __restrict__ W) {
    __shared__ float ldsW[128 * 128];   // 64 KB
    for (int t = threadIdx.x; t < (128 * 128) / 4; t += TB)
        ((float4*)ldsW)[t] = ((const float4*)W)[t];
    __syncthreads();

    const int wave = threadIdx.x >> 5;
    const int lane = threadIdx.x & 31;
    const int rowBase = blockIdx.x * 128 + wave * 16;

    const int mlo = lane & 15;          // row-within-tile (A) / col-within-tile (B)
    const int khalf = lane >> 4;        // which K pair this lane supplies
    const float* aptr = buf + (size_t)(rowBase + mlo) * 128;

    v8f acc[8];
#pragma unroll
    for (int j = 0; j < 8; ++j) acc[j] = (v8f){0.f, 0.f, 0.f, 0.f, 0.f, 0.f, 0.f, 0.f};

    for (int kk = 0; kk < 32; ++kk) {   // K = 128 in steps of 4
        const int kb = kk * 4 + khalf * 2;
        v2f a = *(const v2f*)(aptr + kb);          // one global_load_b64, no guards
#pragma unroll
        for (int j = 0; j < 8; ++j) {   // 8 column tiles of 16 -> N = 128
            v2f b;
            b.x = ldsW[kb * 128 + j * 16 + mlo];
            b.y = ldsW[(kb + 1) * 128 + j * 16 + mlo];
            acc[j] = __builtin_amdgcn_wmma_f32_16x16x4_f32(
                false, a, false, b, (short)0, acc[j], false, false);
        }
    }

    // C/D layout: VGPR r -> row rowBase + r (+8 for lanes 16..31), col = j*16 + (lane&15)
    const int srow = rowBase + (khalf ? 8 : 0);
#pragma unroll
    for (int j = 0; j < 8; ++j) {
#pragma unroll
        for (int r = 0; r < 8; ++r) {
            buf[(size_t)(srow + r) * 128 + j * 16 + mlo] = acc[j][r];
        }
    }
}

// ---------------- edge scatter: agg[dst] += h[src]  (fp32 hardware atomics) ----------------
__global__ void k_scatter(const int* __restrict__ src, const int* __restrict__ dst, int E,
                          const float* __restrict__ h, float* __restrict__ agg) {
    long long t = (long long)blockIdx.x * blockDim.x + threadIdx.x;
    int e = (int)(t >> 5);
    if (e >= E) return;
    int c = ((int)t & 31) * 4;
    int s = src[e], d = dst[e];
    float4 val = *(const float4*)(h + (size_t)s * 128 + c);
    float* p = agg + (size_t)d * 128 + c;
    atomAddF(p + 0, val.x);
    atomAddF(p + 1, val.y);
    atomAddF(p + 2, val.z);
    atomAddF(p + 3, val.w);
}

// out[i][c] = leaky(agg[i][c]*innorm[i] + bias[c])
__global__ void k_finalize(const float* __restrict__ agg, const float* __restrict__ innorm,
                           const float* __restrict__ bias, float* __restrict__ out, int rows) {
    int t = blockIdx.x * blockDim.x + threadIdx.x;
    int i = t >> 5;
    if (i >= rows) return;
    int c = (t & 31) * 4;
    float s = innorm[i];
    float4 g = *(const float4*)(agg + (size_t)i * 128 + c);
    float4 b = *(const float4*)(bias + c);
    float4 r;
    r.x = leaky(g.x * s + b.x);
    r.y = leaky(g.y * s + b.y);
    r.z = leaky(g.z * s + b.z);
    r.w = leaky(g.w * s + b.w);
    *(float4*)(out + (size_t)i * 128 + c) = r;
}

// v[i] = leaky(agg[i]*innorm[i] + b_M) . w_agg + b_agg     (fused, wave32 per row)
__global__ void k_rowdot(const float* __restrict__ agg, const float* __restrict__ innorm,
                         const float* __restrict__ bias, const float* __restrict__ wagg,
                         const float* __restrict__ bagg, float* __restrict__ v, int rows) {
    int t = blockIdx.x * blockDim.x + threadIdx.x;
    int i = t >> 5;
    if (i >= rows) return;
    int lane = t & 31;
    int c = lane * 4;
    float s = innorm[i];
    float4 g = *(const float4*)(agg + (size_t)i * 128 + c);
    float4 b = *(const float4*)(bias + c);
    float4 w = *(const float4*)(wagg + c);
    float p = leaky(g.x * s + b.x) * w.x + leaky(g.y * s + b.y) * w.y +
              leaky(g.z * s + b.z) * w.z + leaky(g.w * s + b.w) * w.w;
#pragma unroll
    for (int off = 16; off; off >>= 1) p += __shfl_down(p, off, 32);
    if (lane == 0) v[i] = p + bagg[0];
}

// t1[j] += sum over row chunk of v[i] * W1[i,j]   (j < 100)
__global__ void k_reduceW1(const float* __restrict__ v, const float* __restrict__ W1,
                           float* __restrict__ t1, int rows) {
    int j = threadIdx.x;
    if (j >= 100) return;
    int start = blockIdx.x * 2048;
    int end = start + 2048; if (end > rows) end = rows;
    float s = 0.f;
    for (int i = start; i < end; ++i) s += v[i] * W1[(size_t)i * 100 + j];
    atomAddF(&t1[j], s);
}

// tiny MLP head: leaky(t1+b1) @ W2 +b2 -> leaky -> @ W3 + b3  -> 10 outputs
__global__ void k_head(const float* __restrict__ t1, const float* __restrict__ b1,
                       const float* __restrict__ W2, const float* __restrict__ b2,
                       const float* __restrict__ W3, const float* __restrict__ b3,
                       float* __restrict__ out) {
    __shared__ float y1[100];
    __shared__ float y2[20];
    int t = threadIdx.x;
    if (t < 100) y1[t] = leaky(t1[t] + b1[t]);
    __syncthreads();
    if (t < 20) {
        float a = b2[t];
        for (int j = 0; j < 100; ++j) a += y1[j] * W2[j * 20 + t];
        y2[t] = leaky(a);
    }
    __syncthreads();
    if (t < 10) {
        float a = b3[t];
        for (int k = 0; k < 20; ++k) a += y2[k] * W3[k * 10 + t];
        out[t] = a;
    }
}

static inline unsigned cdiv(long long a, long long b) { return (unsigned)((a + b - 1) / b); }

extern "C" void kernel_launch(void* const* d_in, const int* in_sizes, int n_in,
                              void* d_out, int out_size, void* d_ws, size_t ws_size,
                              hipStream_t stream) {
    const float* X    = (const float*)d_in[0];
    const int*   src1 = (const int*)d_in[1];
    const int*   dst1 = (const int*)d_in[2];
    const int*   src2 = (const int*)d_in[3];
    const int*   dst2 = (const int*)d_in[4];
    // d_in[5] = M (derived from shapes instead; can't read device scalar under capture)
    const float* W_Q  = (const float*)d_in[6];
    const float* b_Q  = (const float*)d_in[7];
    const float* W_M  = (const float*)d_in[8];
    const float* b_M  = (const float*)d_in[9];
    const float* wagg = (const float*)d_in[10];
    const float* bagg = (const float*)d_in[11];
    const float* W1   = (const float*)d_in[12];
    const float* b1   = (const float*)d_in[13];
    const float* W2   = (const float*)d_in[14];
    const float* b2   = (const float*)d_in[15];
    const float* W3   = (const float*)d_in[16];
    const float* b3   = (const float*)d_in[17];
    float* out = (float*)d_out;

    const int N1 = in_sizes[0] / 128;      // 50000
    const int E1 = in_sizes[1];            // 800000
    const int E2 = in_sizes[3];            // 1600000
    const int N2 = in_sizes[12] / 100;     // M*N1 = 200000
    const int R1 = (N1 + 127) & ~127;      // padded to whole 128-row GEMM blocks
    const int R2 = (N2 + 127) & ~127;

    // workspace layout (floats)
    size_t nA = (size_t)R2 * 128;
    float* A    = (float*)d_ws;            // R2*128
    float* B    = A + nA;                  // R2*128
    float* on1  = B + nA;                  // N1
    float* in1n = on1 + N1;                // N1
    float* on2  = in1n + N1;               // N2
    float* in2n = on2 + N2;                // N2
    float* vvec = in2n + N2;               // N2
    float* t1   = vvec + N2;               // 128
    unsigned* odeg = (unsigned*)(t1 + 128);// N2
    unsigned* ideg = odeg + N2;            // N2

    // ================= Layer 1 (N1 nodes, E1 edges) =================
    k_zero_u32<<<cdiv(2 * (long long)N2, TB), TB, 0, stream>>>(odeg, 2 * N2);
    k_degree<<<cdiv(E1, TB), TB, 0, stream>>>(src1, dst1, E1, odeg, ideg);
    k_norm<<<cdiv(N1, TB), TB, 0, stream>>>(odeg, on1, N1);
    k_norm<<<cdiv(N1, TB), TB, 0, stream>>>(ideg, in1n, N1);

    k_expand_scale<<<cdiv((long long)R1 * 32, TB), TB, 0, stream>>>(X, on1, A, N1, R1, N1);
    k_gemm128_inplace<<<R1 / 128, TB, 0, stream>>>(A, W_Q);            // A = h0 @ W_Q
    k_zero_f32<<<cdiv((long long)N1 * 128, TB), TB, 0, stream>>>(B, N1 * 128);
    k_scatter<<<cdiv((long long)E1 * 32, TB), TB, 0, stream>>>(src1, dst1, E1, A, B);
    k_finalize<<<cdiv((long long)N1 * 32, TB), TB, 0, stream>>>(B, in1n, b_Q, A, N1); // A = h_Q1

    // ================= Layer 2 (N2 nodes, E2 edges) =================
    k_zero_u32<<<cdiv(2 * (long long)N2, TB), TB, 0, stream>>>(odeg, 2 * N2);
    k_degree<<<cdiv(E2, TB), TB, 0, stream>>>(src2, dst2, E2, odeg, ideg);
    k_norm<<<cdiv(N2, TB), TB, 0, stream>>>(odeg, on2, N2);
    k_norm<<<cdiv(N2, TB), TB, 0, stream>>>(ideg, in2n, N2);

    // kron tile + out-norm in one pass: B[i] = A[i % N1] * on2[i]
    k_expand_scale<<<cdiv((long long)R2 * 32, TB), TB, 0, stream>>>(A, on2, B, N2, R2, N1);
    k_gemm128_inplace<<<R2 / 128, TB, 0, stream>>>(B, W_M);            // B = h_M0 @ W_M
    k_zero_f32<<<cdiv((long long)N2 * 128, TB), TB, 0, stream>>>(A, N2 * 128);
    k_scatter<<<cdiv((long long)E2 * 32, TB), TB, 0, stream>>>(src2, dst2, E2, B, A);
    // fused: leaky(norm*agg + b_M) . w_agg + b_agg  -> vvec
    k_rowdot<<<cdiv((long long)N2 * 32, TB), TB, 0, stream>>>(A, in2n, b_M, wagg, bagg, vvec, N2);

    // ================= Head =================
    k_zero_f32<<<1, 128, 0, stream>>>(t1, 128);
    k_reduceW1<<<cdiv(N2, 2048), 128, 0, stream>>>(vvec, W1, t1, N2);
    k_head<<<1, 128, 0, stream>>>(t1, b1, W2, b2, W3, b3, out);
}